// AttnKnn_58892591563305
// MI455X (gfx1250) — compile-verified
//
#include <hip/hip_runtime.h>
#include <math.h>

typedef __attribute__((ext_vector_type(2))) float v2f;
typedef __attribute__((ext_vector_type(8))) float v8f;
typedef int v4i __attribute__((vector_size(4 * sizeof(int))));

#define BB 4
#define CC 64
#define NN 2048
#define KT 50
#define KSEL 20
#define M2 (KT * CC)   // 3200

// dense GEMM block tiling
#define TM 64
#define TN 128
#define TK 32
#define PAD 4          // row padding (floats): keeps 16B alignment, kills bank conflicts

// ---------------------------------------------------------------------------
// Async global->LDS copy of 16 bytes (CDNA5 GLOBAL_LOAD_ASYNC_TO_LDS_B128),
// with a portable fallback. Tracked by ASYNCcnt when the async path is used.
// ---------------------------------------------------------------------------
__device__ __forceinline__ void cp16(float* lds_dst, const float* gsrc) {
#if __has_builtin(__builtin_amdgcn_global_load_async_to_lds_b128)
    __builtin_amdgcn_global_load_async_to_lds_b128(
        (__attribute__((address_space(1))) v4i*)gsrc,
        (__attribute__((address_space(3))) v4i*)lds_dst, 0, 0);
#else
    *(float4*)lds_dst = *(const float4*)gsrc;
#endif
}
__device__ __forceinline__ void wait_async() {
#if __has_builtin(__builtin_amdgcn_global_load_async_to_lds_b128)
    asm volatile("s_wait_asynccnt 0x0" ::: "memory");
#endif
}

// ---------------------------------------------------------------------------
// 1) sq[b,n] = sum_c x[b,c,n]^2
// ---------------------------------------------------------------------------
__global__ void sq_kernel(const float* __restrict__ x, float* __restrict__ sq) {
    long i = (long)blockIdx.x * blockDim.x + threadIdx.x;   // b*N + n
    if (i >= (long)BB * NN) return;
    long b = i / NN, n = i % NN;
    const float* xb = x + b * (long)CC * NN;
    float s = 0.f;
    #pragma unroll 8
    for (int c = 0; c < CC; ++c) { float v = xb[(long)c * NN + n]; s += v * v; }
    sq[i] = s;
}

// ---------------------------------------------------------------------------
// 2) dist[b,n,m] = 2 * sum_c x[c,n] x[c,m] - sq[n] - sq[m]
//    WMMA f32 16x16x4, one wave per 16x16 tile; x fits in L2 (2 MB).
// ---------------------------------------------------------------------------
__global__ __launch_bounds__(32) void gram_dist_kernel(
    const float* __restrict__ x, const float* __restrict__ sq,
    float* __restrict__ dist) {
    const int b = blockIdx.y;
    const int tilesN = NN >> 4;
    const int tm = blockIdx.x / tilesN, tn = blockIdx.x % tilesN;
    const int m0 = tm << 4, n0 = tn << 4;
    const float* xb = x + (long)b * CC * NN;
    const int lane = threadIdx.x;
    const int lmod = lane & 15;
    const int kh = (lane >> 4) << 1;        // 0 or 2
    v8f acc = {};
    #pragma unroll 4
    for (int k = 0; k < CC; k += 4) {
        v2f a, bf;
        a[0]  = xb[(long)(k + kh)     * NN + m0 + lmod];
        a[1]  = xb[(long)(k + kh + 1) * NN + m0 + lmod];
        bf[0] = xb[(long)(k + kh)     * NN + n0 + lmod];
        bf[1] = xb[(long)(k + kh + 1) * NN + n0 + lmod];
        acc = __builtin_amdgcn_wmma_f32_16x16x4_f32(false, a, false, bf,
                                                    (short)0, acc, false, false);
    }
    const float sn = sq[b * NN + n0 + lmod];
    float* db = dist + (long)b * NN * NN;
    const int mh = (lane >> 4) << 3;        // 0 or 8
    #pragma unroll
    for (int i = 0; i < 8; ++i) {
        int m = m0 + i + mh;
        db[(long)m * NN + n0 + lmod] = 2.0f * acc[i] - sq[b * NN + m] - sn;
    }
}

// ---------------------------------------------------------------------------
// 3) per-row top-50 (descending, ties -> lower index), one wave per row.
// ---------------------------------------------------------------------------
__global__ __launch_bounds__(32) void topk50_kernel(
    const float* __restrict__ dist, int* __restrict__ idx50) {
    __shared__ float row[NN];
    const long r = blockIdx.x;              // b*N + n
    const float* d = dist + r * NN;
    const int lane = threadIdx.x;
    for (int i = lane; i < NN; i += 32) row[i] = d[i];
    __syncthreads();
    for (int sel = 0; sel < KT; ++sel) {
        float best = -INFINITY; int bi = NN;
        for (int i = lane; i < NN; i += 32) {
            float v = row[i];
            if (v > best) { best = v; bi = i; }
        }
        #pragma unroll
        for (int off = 16; off > 0; off >>= 1) {
            float ov = __shfl_down(best, off, 32);
            int   oi = __shfl_down(bi,   off, 32);
            if (ov > best || (ov == best && oi < bi)) { best = ov; bi = oi; }
        }
        bi = __shfl(bi, 0, 32);
        if (lane == 0) { idx50[r * KT + sel] = bi; row[bi] = -INFINITY; }
        __syncthreads();
    }
}

// ---------------------------------------------------------------------------
// 4) fused attention: q, kk, logits, softmax, stable 20-smallest selection,
//    v, and pre = attn*v + xwn   (pre laid out as (B, N, 50, 64) flat ==
//    the reference's (B, 3200, 2048) view). One 64-thread block per (b, n).
// ---------------------------------------------------------------------------
__global__ __launch_bounds__(64) void attn_kernel(
    const float* __restrict__ x, const int* __restrict__ idx50,
    const float* __restrict__ Wq, const float* __restrict__ bq,
    const float* __restrict__ Wk, const float* __restrict__ bk,
    const float* __restrict__ Wv, const float* __restrict__ bv,
    float* __restrict__ attn, int* __restrict__ ind20,
    float* __restrict__ pre) {
    __shared__ float qv[CC], xn[CC], xw[CC], red[CC], lg[KT];
    __shared__ int sidx[KT];
    const long r = blockIdx.x;              // b*N + n
    const int b = (int)(r / NN), n = (int)(r % NN);
    const int t = threadIdx.x;
    const float* xb = x + (long)b * CC * NN;
    xn[t] = xb[(long)t * NN + n];
    if (t < KT) sidx[t] = idx50[r * KT + t];
    __syncthreads();
    float q = bq[t];
    #pragma unroll 8
    for (int c = 0; c < CC; ++c) q += xn[c] * Wq[c * CC + t];
    qv[t] = q;
    __syncthreads();
    const float scale = rsqrtf((float)NN);
    for (int k = 0; k < KT; ++k) {
        const int j = sidx[k];
        xw[t] = xb[(long)t * NN + j];
        __syncthreads();
        float kk = bk[t];
        #pragma unroll 8
        for (int c = 0; c < CC; ++c) kk += xw[c] * Wk[c * CC + t];
        red[t] = kk * qv[t];
        __syncthreads();
        #pragma unroll
        for (int s = 32; s > 0; s >>= 1) {
            if (t < s) red[t] += red[t + s];
            __syncthreads();
        }
        if (t == 0) lg[k] = red[0] * scale;
        __syncthreads();
    }
    if (t == 0) {
        float mx = lg[0];
        for (int k = 1; k < KT; ++k) mx = fmaxf(mx, lg[k]);
        float ss = 0.f;
        for (int k = 0; k < KT; ++k) { lg[k] = expf(lg[k] - mx); ss += lg[k]; }
        float inv = 1.0f / ss;
        for (int k = 0; k < KT; ++k) lg[k] *= inv;
        unsigned long long used = 0ull;
        for (int s = 0; s < KSEL; ++s) {
            float bv_ = INFINITY; int bi = 0;
            for (int k = 0; k < KT; ++k) {
                if (used & (1ull << k)) continue;
                if (lg[k] < bv_) { bv_ = lg[k]; bi = k; }
            }
            used |= 1ull << bi;
            ind20[r * KSEL + s] = bi;
        }
    }
    __syncthreads();
    if (t < KT) attn[r * KT + t] = lg[t];
    for (int k = 0; k < KT; ++k) {
        const int j = sidx[k];
        xw[t] = xb[(long)t * NN + j];
        __syncthreads();
        float v = bv[t];
        #pragma unroll 8
        for (int c = 0; c < CC; ++c) v += xw[c] * Wv[c * CC + t];
        pre[(r * KT + k) * CC + t] = lg[k] * v + xw[t];
        __syncthreads();
    }
}

// ---------------------------------------------------------------------------
// 5/7) LayerNorm over last axis (2048 contiguous floats per row).
// ---------------------------------------------------------------------------
__global__ __launch_bounds__(256) void ln_kernel(
    const float* __restrict__ in, float* __restrict__ out,
    const float* __restrict__ g, const float* __restrict__ beta) {
    __shared__ float red[256];
    const long row = blockIdx.x;
    const float* p = in + row * NN;
    float s = 0.f;
    for (int i = threadIdx.x; i < NN; i += 256) s += p[i];
    red[threadIdx.x] = s; __syncthreads();
    #pragma unroll
    for (int st = 128; st > 0; st >>= 1) {
        if (threadIdx.x < st) red[threadIdx.x] += red[threadIdx.x + st];
        __syncthreads();
    }
    const float mu = red[0] * (1.0f / NN);
    __syncthreads();
    float vs = 0.f;
    for (int i = threadIdx.x; i < NN; i += 256) { float d = p[i] - mu; vs += d * d; }
    red[threadIdx.x] = vs; __syncthreads();
    #pragma unroll
    for (int st = 128; st > 0; st >>= 1) {
        if (threadIdx.x < st) red[threadIdx.x] += red[threadIdx.x + st];
        __syncthreads();
    }
    const float rstd = rsqrtf(red[0] * (1.0f / NN) + 1e-5f);
    float* q = out + row * NN;
    for (int i = threadIdx.x; i < NN; i += 256)
        q[i] = (p[i] - mu) * rstd * g[i] + beta[i];
}

// ---------------------------------------------------------------------------
// 6) dense: t2 = o @ Wd + bd + o   (M=3200, N=K=2048 per batch).
//    8-wave workgroup, 64x128 block tile through double-buffered LDS
//    (async global->LDS), each wave computes 32x32 via 2x2 WMMA tiles.
// ---------------------------------------------------------------------------
__global__ __launch_bounds__(256) void dense_wmma_kernel(
    const float* __restrict__ o, const float* __restrict__ Wd,
    const float* __restrict__ bd, float* __restrict__ t2) {
    __shared__ float As[2][TM][TK + PAD];   // [m][k]
    __shared__ float Bs[2][TK][TN + PAD];   // [k][n]
    const int b = blockIdx.z;
    const int tm0 = blockIdx.y * TM;
    const int tn0 = blockIdx.x * TN;
    const float* ob = o + (long)b * M2 * NN;
    float* tb = t2 + (long)b * M2 * NN;
    const int tid = threadIdx.x;
    const int wave = tid >> 5, lane = tid & 31;
    const int wm = wave & 1;                // 0..1 -> M offset wm*32
    const int wn = wave >> 1;               // 0..3 -> N offset wn*32
    const int lmod = lane & 15;
    const int kh = (lane >> 4) << 1;        // 0 or 2
    const int mh = (lane >> 4) << 3;        // 0 or 8
    const int rowA = wm * 32 + lmod;
    const int colB = wn * 32 + lmod;

    // preload chunk 0 into buffer 0
    #pragma unroll
    for (int p = 0; p < 2; ++p) {
        int f = (tid + p * 256) << 2;       // As: 2048 floats
        int m = f >> 5, k = f & 31;
        cp16(&As[0][m][k], &ob[(long)(tm0 + m) * NN + k]);
    }
    #pragma unroll
    for (int p = 0; p < 4; ++p) {
        int f = (tid + p * 256) << 2;       // Bs: 4096 floats
        int kr = f >> 7, n = f & 127;
        cp16(&Bs[0][kr][n], &Wd[(long)kr * NN + tn0 + n]);
    }
    wait_async();
    __syncthreads();

    v8f acc00 = {}, acc01 = {}, acc10 = {}, acc11 = {};
    for (int kc = 0; kc < NN; kc += TK) {
        const int buf = (kc / TK) & 1;
        const int kn = kc + TK;
        if (kn < NN) {                      // issue async loads for next chunk
            const int nbuf = buf ^ 1;
            #pragma unroll
            for (int p = 0; p < 2; ++p) {
                int f = (tid + p * 256) << 2;
                int m = f >> 5, k = f & 31;
                cp16(&As[nbuf][m][k], &ob[(long)(tm0 + m) * NN + kn + k]);
            }
            #pragma unroll
            for (int p = 0; p < 4; ++p) {
                int f = (tid + p * 256) << 2;
                int kr = f >> 7, n = f & 127;
                cp16(&Bs[nbuf][kr][n], &Wd[(long)(kn + kr) * NN + tn0 + n]);
            }
        }
        #pragma unroll
        for (int kk = 0; kk < TK; kk += 4) {
            v2f a0, a1, b0, b1;
            a0[0] = As[buf][rowA][kk + kh];
            a0[1] = As[buf][rowA][kk + kh + 1];
            a1[0] = As[buf][rowA + 16][kk + kh];
            a1[1] = As[buf][rowA + 16][kk + kh + 1];
            b0[0] = Bs[buf][kk + kh][colB];
            b0[1] = Bs[buf][kk + kh + 1][colB];
            b1[0] = Bs[buf][kk + kh][colB + 16];
            b1[1] = Bs[buf][kk + kh + 1][colB + 16];
            acc00 = __builtin_amdgcn_wmma_f32_16x16x4_f32(false, a0, false, b0,
                                                          (short)0, acc00, false, false);
            acc01 = __builtin_amdgcn_wmma_f32_16x16x4_f32(false, a0, false, b1,
                                                          (short)0, acc01, false, false);
            acc10 = __builtin_amdgcn_wmma_f32_16x16x4_f32(false, a1, false, b0,
                                                          (short)0, acc10, false, false);
            acc11 = __builtin_amdgcn_wmma_f32_16x16x4_f32(false, a1, false, b1,
                                                          (short)0, acc11, false, false);
        }
        wait_async();
        __syncthreads();
    }

    const float bn0 = bd[tn0 + colB];
    const float bn1 = bd[tn0 + colB + 16];
    #pragma unroll
    for (int i = 0; i < 8; ++i) {
        const int ma = tm0 + wm * 32 + i + mh;
        const int mb = ma + 16;
        const int na = tn0 + colB;
        const int nb = na + 16;
        tb[(long)ma * NN + na] = acc00[i] + bn0 + ob[(long)ma * NN + na];
        tb[(long)ma * NN + nb] = acc01[i] + bn1 + ob[(long)ma * NN + nb];
        tb[(long)mb * NN + na] = acc10[i] + bn0 + ob[(long)mb * NN + na];
        tb[(long)mb * NN + nb] = acc11[i] + bn1 + ob[(long)mb * NN + nb];
    }
}

// ---------------------------------------------------------------------------
// 8) final FC (only the 20 selected neighbors) + residual + gather/transpose.
// ---------------------------------------------------------------------------
__global__ __launch_bounds__(64) void fc_gather_kernel(
    const float* __restrict__ x2, const float* __restrict__ x,
    const int* __restrict__ idx50, const int* __restrict__ ind20,
    const float* __restrict__ Wfc, const float* __restrict__ bfc,
    float* __restrict__ out) {
    __shared__ float vec[CC];
    __shared__ int s20[KSEL], s50[KT];
    const long r = blockIdx.x;              // b*N + n
    const int b = (int)(r / NN), n = (int)(r % NN);
    const int t = threadIdx.x;
    const float* xb = x + (long)b * CC * NN;
    const float* x2b = x2 + (long)b * M2 * NN;
    if (t < KSEL) s20[t] = ind20[r * KSEL + t];
    if (t < KT)   s50[t] = idx50[r * KT + t];
    __syncthreads();
    for (int s = 0; s < KSEL; ++s) {
        const int k = s20[s];
        const int j = s50[k];
        vec[t] = x2b[(long)(k * CC + t) * NN + n] + xb[(long)t * NN + j];
        __syncthreads();
        float acc = bfc[t];
        #pragma unroll 8
        for (int c = 0; c < CC; ++c) acc += vec[c] * Wfc[c * CC + t];
        out[(((long)b * CC + t) * KSEL + s) * NN + n] = acc;
        __syncthreads();
    }
}

// ---------------------------------------------------------------------------
extern "C" void kernel_launch(void* const* d_in, const int* in_sizes, int n_in,
                              void* d_out, int out_size, void* d_ws, size_t ws_size,
                              hipStream_t stream) {
    (void)in_sizes; (void)n_in; (void)out_size; (void)ws_size;
    const float* x   = (const float*)d_in[0];
    const float* Wq  = (const float*)d_in[1];
    const float* bq  = (const float*)d_in[2];
    const float* Wk  = (const float*)d_in[3];
    const float* bk  = (const float*)d_in[4];
    const float* Wv  = (const float*)d_in[5];
    const float* bv  = (const float*)d_in[6];
    const float* Wd  = (const float*)d_in[7];
    const float* bd  = (const float*)d_in[8];
    const float* Wfc = (const float*)d_in[9];
    const float* bfc = (const float*)d_in[10];
    const float* g1  = (const float*)d_in[11];
    const float* be1 = (const float*)d_in[12];
    const float* g2  = (const float*)d_in[13];
    const float* be2 = (const float*)d_in[14];
    float* out = (float*)d_out;

    char* ws = (char*)d_ws;
    size_t off = 0;
    auto take = [&](size_t nbytes) -> char* {
        char* p = ws + off;
        off = (off + nbytes + 255) & ~(size_t)255;
        return p;
    };
    float* sq    = (float*)take((size_t)BB * NN * 4);
    float* dist  = (float*)take((size_t)BB * NN * NN * 4);
    int*   idx50 = (int*)  take((size_t)BB * NN * KT * 4);
    float* attn  = (float*)take((size_t)BB * NN * KT * 4);
    int*   ind20 = (int*)  take((size_t)BB * NN * KSEL * 4);
    float* bufA  = (float*)take((size_t)BB * M2 * NN * 4);  // pre -> t2
    float* bufB  = (float*)take((size_t)BB * M2 * NN * 4);  // o   -> x2

    sq_kernel<<<(BB * NN + 255) / 256, 256, 0, stream>>>(x, sq);

    dim3 g2d((NN / 16) * (NN / 16), BB);
    gram_dist_kernel<<<g2d, 32, 0, stream>>>(x, sq, dist);

    topk50_kernel<<<BB * NN, 32, 0, stream>>>(dist, idx50);

    attn_kernel<<<BB * NN, 64, 0, stream>>>(x, idx50, Wq, bq, Wk, bk, Wv, bv,
                                            attn, ind20, bufA);

    ln_kernel<<<BB * M2, 256, 0, stream>>>(bufA, bufB, g1, be1);

    dim3 gdd(NN / TN, M2 / TM, BB);
    dense_wmma_kernel<<<gdd, 256, 0, stream>>>(bufB, Wd, bd, bufA);

    ln_kernel<<<BB * M2, 256, 0, stream>>>(bufA, bufB, g2, be2);

    fc_gather_kernel<<<BB * NN, 64, 0, stream>>>(bufB, x, idx50, ind20,
                                                 Wfc, bfc, out);
}